// SourceBasedSCNLayer_34565896798994
// MI455X (gfx1250) — compile-verified
//
#include <hip/hip_runtime.h>

typedef __attribute__((ext_vector_type(2))) float v2f;
typedef __attribute__((ext_vector_type(8))) float v8f;

#define DD 128
#define LN_EPS 1e-5f

// ---------------------------------------------------------------------------
// Kernel 1: COO SpMM scatter.  One wave32 per edge.
//   support[row, :] += val * x[col, :]
// Each lane handles a float4 (32 lanes * 4 = 128 = D), 4 fp32 HW atomics.
// All traffic stays resident in the 192MB L2 -> atomic-bandwidth bound.
// ---------------------------------------------------------------------------
__global__ __launch_bounds__(256)
void spmm_scatter(const float* __restrict__ x,
                  const float* __restrict__ edge_val,
                  const int*   __restrict__ edge_row,
                  const int*   __restrict__ edge_col,
                  float*       __restrict__ support,
                  int E)
{
    const int wave = (blockIdx.x * blockDim.x + threadIdx.x) >> 5;
    const int lane = threadIdx.x & 31;
    if (wave >= E) return;

    const int   r = edge_row[wave];
    const int   c = edge_col[wave];
    const float v = edge_val[wave];

    const float4 xv = ((const float4*)(x + (size_t)c * DD))[lane];
    float* dst = support + (size_t)r * DD + lane * 4;

    __hip_atomic_fetch_add(dst + 0, v * xv.x, __ATOMIC_RELAXED, __HIP_MEMORY_SCOPE_AGENT);
    __hip_atomic_fetch_add(dst + 1, v * xv.y, __ATOMIC_RELAXED, __HIP_MEMORY_SCOPE_AGENT);
    __hip_atomic_fetch_add(dst + 2, v * xv.z, __ATOMIC_RELAXED, __HIP_MEMORY_SCOPE_AGENT);
    __hip_atomic_fetch_add(dst + 3, v * xv.w, __ATOMIC_RELAXED, __HIP_MEMORY_SCOPE_AGENT);
}

// ---------------------------------------------------------------------------
// Kernel 2: fused  h = support @ W^T + b ; LayerNorm(h) ; ReLU
// One workgroup (8 wave32 = 256 threads) per 16-row stripe.
// Wave w computes the 16x16 output tile for columns [16w, 16w+16) via
// 32 x V_WMMA_F32_16X16X4_F32 (K=128, fp32-exact), stages h in LDS,
// then a cross-wave LDS reduction does the LayerNorm.
//
// A-matrix 16x4 f32 layout (ISA 7.12.2): lane l holds M=(l&15),
//   VGPR0 = K = 2*(l>>4), VGPR1 = K = 2*(l>>4)+1  (per 4-wide K step).
// B-matrix 4x16: lane l holds N=(l&15), same K split.  B[k][n] = W[n][k].
// C/D 16x16 f32: VGPR j = (M = j + 8*(l>>4), N = l&15).
// ---------------------------------------------------------------------------
__global__ __launch_bounds__(256)
void gemm_ln_relu(const float* __restrict__ support,
                  const float* __restrict__ W,       // [Dout, Din] row-major
                  const float* __restrict__ bias,
                  const float* __restrict__ gamma,
                  const float* __restrict__ beta,
                  float*       __restrict__ out,
                  int N)
{
    __shared__ float h[16][DD + 4];      // +4 pad: row stride 132 words (banks=64)
    __shared__ float red_s[16][8];
    __shared__ float red_q[16][8];
    __shared__ float mu_s[16];
    __shared__ float rs_s[16];

    const int tid  = threadIdx.x;
    const int lane = tid & 31;
    const int wave = tid >> 5;           // 0..7 : column tile
    const int half = lane >> 4;          // 0 or 1
    const int l16  = lane & 15;
    const int m0   = blockIdx.x * 16;

    // clamped row for tail safety (N=50000 is a multiple of 16, but be safe)
    const int mrow = min(m0 + l16, N - 1);

    const float* Aptr = support + (size_t)mrow * DD + 2 * half;
    const float* Bptr = W + (size_t)(wave * 16 + l16) * DD + 2 * half;

    v8f acc = {};
    #pragma unroll
    for (int kk = 0; kk < 32; ++kk) {
        const v2f a = *(const v2f*)(Aptr + kk * 4);
        const v2f b = *(const v2f*)(Bptr + kk * 4);
        acc = __builtin_amdgcn_wmma_f32_16x16x4_f32(
            /*neg_a=*/false, a, /*neg_b=*/false, b,
            /*c_mod=*/(short)0, acc, /*reuse_a=*/false, /*reuse_b=*/false);
    }

    // stage h (+bias) into LDS
    {
        const int col = wave * 16 + l16;
        const float bb = bias[col];
        #pragma unroll
        for (int j = 0; j < 8; ++j) {
            const int row = j + 8 * half;
            h[row][col] = acc[j] + bb;
        }
    }
    __syncthreads();

    // per-row partial sums: 128 threads, 8 segments of 16 cols per row
    if (tid < 128) {
        const int row = tid >> 3;
        const int seg = tid & 7;
        float s = 0.f, q = 0.f;
        #pragma unroll
        for (int c = 0; c < 16; ++c) {
            const float v = h[row][seg * 16 + c];
            s += v;
            q += v * v;
        }
        red_s[row][seg] = s;
        red_q[row][seg] = q;
    }
    __syncthreads();

    if (tid < 16) {
        float s = 0.f, q = 0.f;
        #pragma unroll
        for (int j = 0; j < 8; ++j) { s += red_s[tid][j]; q += red_q[tid][j]; }
        const float mu  = s * (1.0f / DD);
        const float var = q * (1.0f / DD) - mu * mu;
        mu_s[tid] = mu;
        rs_s[tid] = __frsqrt_rn(var + LN_EPS);
    }
    __syncthreads();

    // normalize + affine + ReLU, 2048 elements / 256 threads = 8 each
    #pragma unroll
    for (int i = 0; i < 8; ++i) {
        const int idx = tid + 256 * i;
        const int row = idx >> 7;
        const int col = idx & (DD - 1);
        const int gr  = m0 + row;
        if (gr < N) {
            const float v = (h[row][col] - mu_s[row]) * rs_s[row] * gamma[col] + beta[col];
            out[(size_t)gr * DD + col] = fmaxf(v, 0.f);
        }
    }
}

// ---------------------------------------------------------------------------
// Inputs (setup_inputs order): x, edge_val, W, b, gamma, beta, edge_row, edge_col
// ---------------------------------------------------------------------------
extern "C" void kernel_launch(void* const* d_in, const int* in_sizes, int n_in,
                              void* d_out, int out_size, void* d_ws, size_t ws_size,
                              hipStream_t stream)
{
    const float* x        = (const float*)d_in[0];
    const float* edge_val = (const float*)d_in[1];
    const float* W        = (const float*)d_in[2];
    const float* b        = (const float*)d_in[3];
    const float* gamma    = (const float*)d_in[4];
    const float* beta     = (const float*)d_in[5];
    const int*   edge_row = (const int*)d_in[6];
    const int*   edge_col = (const int*)d_in[7];
    float*       out      = (float*)d_out;

    const int N = in_sizes[0] / DD;
    const int E = in_sizes[1];

    float* support = (float*)d_ws;                 // [N, D] fp32 scratch
    hipMemsetAsync(support, 0, (size_t)N * DD * sizeof(float), stream);

    // 8 edges (waves) per 256-thread block
    const int eblocks = (E + 7) / 8;
    spmm_scatter<<<eblocks, 256, 0, stream>>>(x, edge_val, edge_row, edge_col,
                                              support, E);

    const int mblocks = (N + 15) / 16;
    gemm_ln_relu<<<mblocks, 256, 0, stream>>>(support, W, b, gamma, beta, out, N);
}